// SimpleSphereTracer_46840913330440
// MI455X (gfx1250) — compile-verified
//
#include <hip/hip_runtime.h>

// ---------------------------------------------------------------------------
// SimpleSphereTracer on MI455X (gfx1250, wave32).
//
// Exact algebraic collapse: feat has no activation, so
//   (q@W1+b1)@W2+b2 == q@(W1@W2) + (b1@W2+b2)
// -> each step is a 3->16 linear (K padded to 4), one V_WMMA_F32_16X16X4_F32
// per 16-point tile. The GEMM is done TRANSPOSED (D^T = M^T @ q^T) so that a
// point's features land in one lane's accumulator VGPRs: LayerNorm / ReLU /
// SDF-dot reduce in-lane over the v8f plus a single xor-16 ds_swizzle per
// reduction (3 swizzles per step total, vs 96 bpermutes in the row-major
// mapping). Memory-bound overall: ~15 MB traffic.
// ---------------------------------------------------------------------------

typedef float v2f __attribute__((ext_vector_type(2)));
typedef float v8f __attribute__((ext_vector_type(8)));

#define MIN_DEPTH 0.05f
#define LN_EPS 1e-5f
#define STEPS 4
#define WAVES_PER_BLOCK 8
#define SWAPX16 0x401F  // ds_swizzle group-of-32: xor_mask=0x10, and_mask=0x1f

__device__ __forceinline__ float xor16_add(float v) {
  // v + (v from lane^16), single ds_swizzle_b32 with immediate pattern
  int o = __builtin_amdgcn_ds_swizzle(__float_as_int(v), SWAPX16);
  return v + __int_as_float(o);
}

// ws layout (floats): [0..47] M[k][j] = (W1@W2)[k,j] (k<3, j<16, row-major)
//                     [48..63] c[j]   = (b1@W2 + b2)[j]
__global__ void precompute_combined(const float* __restrict__ feat_W,  // [3,256]
                                    const float* __restrict__ feat_b,  // [256]
                                    const float* __restrict__ ffn_W,   // [256,16]
                                    const float* __restrict__ ffn_b,   // [16]
                                    float* __restrict__ ws) {
  const int tid = threadIdx.x;
  if (tid < 48) {
    const int k = tid >> 4, j = tid & 15;
    float acc = 0.f;
    for (int f = 0; f < 256; ++f)
      acc = fmaf(feat_W[k * 256 + f], ffn_W[f * 16 + j], acc);
    ws[tid] = acc;
  } else if (tid < 64) {
    const int j = tid - 48;
    float acc = ffn_b[j];
    for (int f = 0; f < 256; ++f)
      acc = fmaf(feat_b[f], ffn_W[f * 16 + j], acc);
    ws[48 + j] = acc;
  }
}

__global__ void __launch_bounds__(32 * WAVES_PER_BLOCK)
sphere_trace_wmma(const float* __restrict__ start,   // [N,3]
                  const float* __restrict__ rdir,    // [N,3]
                  const float* __restrict__ Mws,     // combined weights (d_ws)
                  const float* __restrict__ ln_g,    // [16]
                  const float* __restrict__ ln_b,    // [16]
                  const float* __restrict__ sdf_W,   // [16]
                  const float* __restrict__ sdf_b,   // [1]
                  float* __restrict__ out,           // [N]
                  int ntiles) {
  const int lane = threadIdx.x & 31;
  const int wave = threadIdx.x >> 5;
  const int tile = blockIdx.x * WAVES_PER_BLOCK + wave;
  if (tile >= ntiles) return;  // wave-uniform: EXEC stays all-1s for WMMA

  const bool lo = lane < 16;
  const int m = lane & 15;      // A row = feature m ; B col = point m
  const int fsel = lo ? 0 : 8;  // which 8 features this half-wave owns in C/D

  // --- A operand: M^T (16 features x K=4, K=3 padded). Step-invariant. ---
  // A layout: lanes 0-15 -> (K0,K1) in (v0,v1); lanes 16-31 -> (K2,K3).
  v2f amat;
  amat.x = Mws[(lo ? 0 : 2) * 16 + m];   // M[0][m] / M[2][m]
  amat.y = lo ? Mws[1 * 16 + m] : 0.f;   // M[1][m] / K=3 zero pad

  // --- per-half-wave feature-bank constants (features fsel..fsel+7) -----
  float cv[8], glv[8], blv[8], swv[8];
#pragma unroll
  for (int r = 0; r < 8; ++r) {
    cv[r]  = Mws[48 + fsel + r];  // combined bias
    glv[r] = ln_g[fsel + r];
    blv[r] = ln_b[fsel + r];
    swv[r] = sdf_W[fsel + r];
  }
  const float sb = sdf_b[0];

  // --- ray data for point m (B operand layout: lo=(x,y), hi=(z,0)) ------
  const size_t pbase = ((size_t)tile * 16 + m) * 3;
  const float s0 = lo ? start[pbase + 0] : start[pbase + 2];
  const float v0 = lo ? rdir[pbase + 0] : rdir[pbase + 2];
  const float s1 = lo ? start[pbase + 1] : 0.f;
  const float v1 = lo ? rdir[pbase + 1] : 0.f;

  // depth for point m, replicated in both half-waves
  float depth = MIN_DEPTH;

#pragma unroll
  for (int step = 0; step < STEPS; ++step) {
    // B operand: query^T = (start + dir*depth)^T, K=3 padded to 4
    v2f bmat;
    bmat.x = fmaf(v0, depth, s0);
    bmat.y = fmaf(v1, depth, s1);   // exactly 0 in hi half (K=3 pad)

    v8f acc;
#pragma unroll
    for (int i = 0; i < 8; ++i) acc[i] = cv[i];  // bias via C operand

    // D[feat][pt] = sum_k M[k][feat]*query[pt][k] + c[feat]
    // lane n, VGPR r  ->  h_pre[point n][feature fsel+r]
    v8f dmat = __builtin_amdgcn_wmma_f32_16x16x4_f32(
        /*neg_a=*/false, amat, /*neg_b=*/false, bmat,
        /*c_mod=*/(short)0, acc, /*reuse_a=*/false, /*reuse_b=*/false);

    // ---- LayerNorm over 16 features: 8 in-lane + 8 in lane^16 ----------
    float s = dmat[0];
#pragma unroll
    for (int r = 1; r < 8; ++r) s += dmat[r];
    const float mu = xor16_add(s) * 0.0625f;

    float q = 0.f;
    float dx[8];
#pragma unroll
    for (int r = 0; r < 8; ++r) {
      dx[r] = dmat[r] - mu;
      q = fmaf(dx[r], dx[r], q);
    }
    const float var = xor16_add(q) * 0.0625f;
    const float inv = rsqrtf(var + LN_EPS);

    // affine + ReLU + SDF-head dot (local 8 features)
    float t = 0.f;
#pragma unroll
    for (int r = 0; r < 8; ++r) {
      float h = fmaf(dx[r] * inv, glv[r], blv[r]);
      h = fmaxf(h, 0.f);
      t = fmaf(h, swv[r], t);
    }
    // xor-16 combine gives the full 16-feature dot to BOTH halves, already
    // replicated per point (lane&15) -- exactly depth's layout.
    depth += xor16_add(t) + sb;
  }

  if (lo) out[(size_t)tile * 16 + m] = depth;
}

extern "C" void kernel_launch(void* const* d_in, const int* in_sizes, int n_in,
                              void* d_out, int out_size, void* d_ws, size_t ws_size,
                              hipStream_t stream) {
  const float* start  = (const float*)d_in[0];
  const float* rdir   = (const float*)d_in[1];
  const float* feat_W = (const float*)d_in[2];
  const float* feat_b = (const float*)d_in[3];
  const float* ffn_W  = (const float*)d_in[4];
  const float* ffn_b  = (const float*)d_in[5];
  const float* ln_g   = (const float*)d_in[6];
  const float* ln_bb  = (const float*)d_in[7];
  const float* sdf_W  = (const float*)d_in[8];
  const float* sdf_b  = (const float*)d_in[9];
  float* out = (float*)d_out;
  float* ws  = (float*)d_ws;  // 64 floats used

  // Fold the 256-wide linear into a 3x16 matrix + 16-bias (same stream ->
  // ordered before the tracer).
  precompute_combined<<<1, 64, 0, stream>>>(feat_W, feat_b, ffn_W, ffn_b, ws);

  const int npts = in_sizes[0] / 3;            // 524288
  const int ntiles = npts / 16;                // 32768 (N is a multiple of 16)
  const int nblocks = (ntiles + WAVES_PER_BLOCK - 1) / WAVES_PER_BLOCK;
  sphere_trace_wmma<<<nblocks, 32 * WAVES_PER_BLOCK, 0, stream>>>(
      start, rdir, ws, ln_g, ln_bb, sdf_W, sdf_b, out, ntiles);
}